// SparseResNetEncoder_5471788335529
// MI455X (gfx1250) — compile-verified
//
#include <hip/hip_runtime.h>

// CDNA5 / gfx1250, wave32. Sparse conv chain via V_WMMA_F32_16X16X4_F32.
// fp32 tensor op keeps full precision vs the fp32 reference while using the
// matrix pipe. Each wave computes RT (=2) 16x16 output tiles sharing the same
// B (weight) registers: two independent accumulator chains hide the XDL
// WMMA->WMMA RAW latency, and B vmem traffic is halved.

typedef __attribute__((ext_vector_type(2))) float v2f;
typedef __attribute__((ext_vector_type(8))) float v8f;

template <int CIN, int COUT, int RT>
__global__ __launch_bounds__(256) void spconv_wmma_kernel(
    const float* __restrict__ x,     // (n_in, CIN) gathered via nmap
    const int* __restrict__ nmap,    // (27, n_out), -1 = missing neighbor
    const float* __restrict__ W,     // (27, CIN, COUT)
    float* __restrict__ y,           // (n_out, COUT)
    int n_out, int n_wave_tiles) {
  const int lane = threadIdx.x & 31;
  const int wave = threadIdx.x >> 5;
  const int wt = blockIdx.x * 8 + wave;  // wave tile: RT*16 rows (uniform per wave)
  if (wt >= n_wave_tiles) return;        // wave-uniform exit: EXEC all-1s below
  const int tn = blockIdx.y;             // 16-col output tile

  const int half = lane >> 4;  // 0 or 1
  const int l16 = lane & 15;
  const int col = (tn << 4) + l16;  // output col this lane owns (B/C side)
  const int khalf = half << 1;      // K pair base: 0 or 2 within a K=4 step
  const int m0 = wt * (16 * RT) + l16;

  v8f acc[RT] = {};

#pragma unroll 1
  for (int k = 0; k < 27; ++k) {
    const int* __restrict__ nmk = nmap + (size_t)k * n_out;
    int idx[RT];
#pragma unroll
    for (int r = 0; r < RT; ++r) {
      const int m = m0 + 16 * r;
      idx[r] = (m < n_out) ? nmk[m] : -1;
    }
    // nmap address stream is predictable one k ahead -> global_prefetch_b8.
    if (k + 1 < 27 && m0 < n_out) __builtin_prefetch(nmk + n_out + m0, 0, 1);

    const float* __restrict__ Wk = W + (size_t)k * (CIN * COUT);
#pragma unroll
    for (int kb = 0; kb < CIN; kb += 4) {
      const int k0 = kb + khalf;
      // B 4x16 f32 layout: lane<16 -> K={kb,kb+1}, lane>=16 -> K={kb+2,kb+3},
      // col striped across lanes 0..15. Loaded once, used by RT WMMAs.
      v2f b;
      if constexpr (CIN >= 4) {
        b.x = Wk[(size_t)k0 * COUT + col];
        b.y = Wk[(size_t)(k0 + 1) * COUT + col];
      } else {  // CIN==1 (layer 1): zero-pad K up to 4
        b.x = (k0 < CIN) ? Wk[(size_t)k0 * COUT + col] : 0.0f;
        b.y = (k0 + 1 < CIN) ? Wk[(size_t)(k0 + 1) * COUT + col] : 0.0f;
      }
#pragma unroll
      for (int r = 0; r < RT; ++r) {
        const bool gok = (idx[r] >= 0);
        const float* __restrict__ xr = x + (size_t)(gok ? idx[r] : 0) * CIN;
        // A 16x4 f32 layout: row = lane&15, K pair selected by lane>>4.
        v2f a;
        if constexpr (CIN >= 4) {
          a = gok ? *(const v2f*)(xr + k0) : (v2f){0.0f, 0.0f};
        } else {
          a.x = (gok && k0 < CIN) ? xr[k0] : 0.0f;
          a.y = (gok && k0 + 1 < CIN) ? xr[k0 + 1] : 0.0f;
        }
        acc[r] = __builtin_amdgcn_wmma_f32_16x16x4_f32(
            /*neg_a=*/false, a, /*neg_b=*/false, b,
            /*c_mod=*/(short)0, acc[r], /*reuse_a=*/false, /*reuse_b=*/false);
      }
    }
  }

  // C/D layout: VGPR j holds row (j + 8*half), col = l16.
#pragma unroll
  for (int r = 0; r < RT; ++r) {
    const int row_base = wt * (16 * RT) + 16 * r + (half << 3);
#pragma unroll
    for (int j = 0; j < 8; ++j) {
      const int row = row_base + j;
      if (row < n_out) y[(size_t)row * COUT + col] = acc[r][j];
    }
  }
}

template <int CIN, int COUT>
static void launch_layer(const float* x, const int* nmap, const float* W,
                         float* y, int n_out, hipStream_t s) {
  if (n_out <= 0) return;
  constexpr int RT = 2;  // row tiles per wave
  const int n_wave_tiles = (n_out + 16 * RT - 1) / (16 * RT);
  dim3 grid((unsigned)((n_wave_tiles + 7) / 8), (unsigned)(COUT / 16), 1);
  spconv_wmma_kernel<CIN, COUT, RT>
      <<<grid, dim3(256, 1, 1), 0, s>>>(x, nmap, W, y, n_out, n_wave_tiles);
}

static inline size_t align_up(size_t v, size_t a) { return (v + a - 1) & ~(a - 1); }

extern "C" void kernel_launch(void* const* d_in, const int* in_sizes, int n_in,
                              void* d_out, int out_size, void* d_ws, size_t ws_size,
                              hipStream_t stream) {
  (void)n_in; (void)out_size; (void)ws_size;

  // setup_inputs() dict order: feats, W1, nmap1, W2, nmap2, ..., W5, nmap5
  const float* feats = (const float*)d_in[0];
  const float* W1 = (const float*)d_in[1];
  const int*   nm1 = (const int*)d_in[2];
  const float* W2 = (const float*)d_in[3];
  const int*   nm2 = (const int*)d_in[4];
  const float* W3 = (const float*)d_in[5];
  const int*   nm3 = (const int*)d_in[6];
  const float* W4 = (const float*)d_in[7];
  const int*   nm4 = (const int*)d_in[8];
  const float* W5 = (const float*)d_in[9];
  const int*   nm5 = (const int*)d_in[10];

  const int n1 = in_sizes[2] / 27;
  const int n2 = in_sizes[4] / 27;
  const int n3 = in_sizes[6] / 27;
  const int n4 = in_sizes[8] / 27;
  const int n5 = in_sizes[10] / 27;

  // Carve fp32 intermediates out of the workspace.
  char* ws = (char*)d_ws;
  float* x1 = (float*)ws; ws += align_up((size_t)n1 * 16 * sizeof(float), 256);
  float* x2 = (float*)ws; ws += align_up((size_t)n2 * 32 * sizeof(float), 256);
  float* x3 = (float*)ws; ws += align_up((size_t)n3 * 64 * sizeof(float), 256);
  float* x4 = (float*)ws;
  float* out = (float*)d_out;

  launch_layer<1, 16>(feats, nm1, W1, x1, n1, stream);
  launch_layer<16, 32>(x1, nm2, W2, x2, n2, stream);
  launch_layer<32, 64>(x2, nm3, W3, x3, n3, stream);
  launch_layer<64, 128>(x3, nm4, W4, x4, n4, stream);
  launch_layer<128, 512>(x4, nm5, W5, out, n5, stream);
}